// GCN_31439160607360
// MI455X (gfx1250) — compile-verified
//
#include <hip/hip_runtime.h>
#include <hip/hip_bf16.h>
#include <math.h>

typedef __attribute__((ext_vector_type(2))) float v2f;
typedef __attribute__((ext_vector_type(8))) float v8f;

namespace {
constexpr int Nn  = 200000;   // total nodes
constexpr int NPG = 12500;    // nodes per graph
constexpr int Bg  = 16;       // graphs
constexpr int Eg  = 1600000;  // edges
constexpr int K1  = 1250;     // SAGPool1 keep per graph
constexpr int N2  = Bg * K1;  // 20000 pooled nodes
constexpr float BN_EPS = 1e-5f;

__device__ __forceinline__ unsigned f2u(float f) {
  unsigned u = __float_as_uint(f);
  return (u & 0x80000000u) ? ~u : (u | 0x80000000u);
}

// ---------------- generic fills ----------------
__global__ void fill_f(float* p, float v, size_t n) {
  size_t i = (size_t)blockIdx.x * blockDim.x + threadIdx.x;
  if (i < n) p[i] = v;
}
__global__ void fill_i(int* p, int v, size_t n) {
  size_t i = (size_t)blockIdx.x * blockDim.x + threadIdx.x;
  if (i < n) p[i] = v;
}

// ---------------- degree / norm ----------------
__global__ void deg_acc(const int* __restrict__ srcv, const int* __restrict__ dstv,
                        float* __restrict__ deg, int nE) {
  int e = blockIdx.x * blockDim.x + threadIdx.x;
  if (e >= nE) return;
  int s = srcv[e], d = dstv[e];
  if (s >= 0 && d >= 0) atomicAdd(&deg[d], 1.0f);
}
__global__ void rsq_k(const float* __restrict__ deg, float* __restrict__ dinv, int n) {
  int i = blockIdx.x * blockDim.x + threadIdx.x;
  if (i < n) dinv[i] = rsqrtf(deg[i]);
}

// ---------------- WMMA fp32 GEMM: Y[M x NOUT] = X[M x K] @ W[K x NOUT] ----------------
// One wave computes one 16x16 tile via V_WMMA_F32_16X16X4_F32.  M % 16 == 0.
template <int K, int NOUT>
__global__ __launch_bounds__(256) void gemm_wmma(const float* __restrict__ X,
                                                 const float* __restrict__ W,
                                                 float* __restrict__ Y, int M) {
  constexpr int TN = NOUT / 16;
  constexpr int KP = (K + 3) & ~3;
  int wave = (int)((blockIdx.x * blockDim.x + threadIdx.x) >> 5);  // wave-uniform
  int lane = threadIdx.x & 31;
  int tiles = (M / 16) * TN;
  if (wave >= tiles) return;                // uniform per wave -> EXEC stays full
  int tm = (wave / TN) * 16;
  int tn = (wave % TN) * 16;
  int half = lane >> 4;                     // 0: K+0/K+1, 1: K+2/K+3
  int r = lane & 15;
  v8f c = {};
  for (int k0 = 0; k0 < KP; k0 += 4) {
    int ka = k0 + 2 * half;
    v2f a, b;
    a.x = (ka     < K) ? X[(size_t)(tm + r) * K + ka]     : 0.0f;
    a.y = (ka + 1 < K) ? X[(size_t)(tm + r) * K + ka + 1] : 0.0f;
    b.x = (ka     < K) ? W[(size_t)ka * NOUT + tn + r]       : 0.0f;
    b.y = (ka + 1 < K) ? W[(size_t)(ka + 1) * NOUT + tn + r] : 0.0f;
    c = __builtin_amdgcn_wmma_f32_16x16x4_f32(false, a, false, b, (short)0, c,
                                              false, false);
  }
#pragma unroll
  for (int v = 0; v < 8; ++v)
    Y[(size_t)(tm + v + 8 * half) * NOUT + tn + r] = c[v];
}

// ---------------- edge aggregation: A[d] += G[s] * dinv[s]*dinv[d] ----------------
template <int F>
__global__ void edge_agg(const float* __restrict__ G, const int* __restrict__ srcv,
                         const int* __restrict__ dstv, const float* __restrict__ dinv,
                         float* __restrict__ A, int nE) {
  size_t idx = (size_t)blockIdx.x * blockDim.x + threadIdx.x;
  if (idx >= (size_t)nE * F) return;
  int e = (int)(idx / F), f = (int)(idx % F);
  int s = srcv[e], d = dstv[e];
  if (s < 0 || d < 0) return;               // masked edge (em == 0)
  float nrm = dinv[s] * dinv[d];
  atomicAdd(&A[(size_t)d * F + f], G[(size_t)s * F + f] * nrm);
}

// ---------------- self-loop + bias: A += G * dinv^2 + b ----------------
template <int F>
__global__ void selfloop_bias(float* __restrict__ A, const float* __restrict__ G,
                              const float* __restrict__ dinv, const float* __restrict__ b,
                              int n) {
  size_t idx = (size_t)blockIdx.x * blockDim.x + threadIdx.x;
  if (idx >= (size_t)n * F) return;
  int v = (int)(idx / F), f = (int)(idx % F);
  float di = dinv[v];
  A[idx] += G[idx] * di * di + b[f];
}

// ---------------- BatchNorm stats (sum, sumsq per channel) ----------------
template <int F>
__global__ void bn_stats(const float* __restrict__ x, float* __restrict__ sums,
                         float* __restrict__ sqs, int n) {
  __shared__ float ssum[F];
  __shared__ float ssq[F];
  if (threadIdx.x < F) { ssum[threadIdx.x] = 0.0f; ssq[threadIdx.x] = 0.0f; }
  __syncthreads();
  size_t total = (size_t)n * F;
  for (size_t i = (size_t)blockIdx.x * blockDim.x + threadIdx.x; i < total;
       i += (size_t)gridDim.x * blockDim.x) {
    float v = x[i];
    int f = (int)(i % F);
    atomicAdd(&ssum[f], v);
    atomicAdd(&ssq[f], v * v);
  }
  __syncthreads();
  if (threadIdx.x < F) {
    atomicAdd(&sums[threadIdx.x], ssum[threadIdx.x]);
    atomicAdd(&sqs[threadIdx.x], ssq[threadIdx.x]);
  }
}

template <int F>
__global__ void bn_apply_tanh(float* __restrict__ x, const float* __restrict__ sums,
                              const float* __restrict__ sqs, const float* __restrict__ gam,
                              const float* __restrict__ bet, int n) {
  size_t idx = (size_t)blockIdx.x * blockDim.x + threadIdx.x;
  if (idx >= (size_t)n * F) return;
  int f = (int)(idx % F);
  float inv_n = 1.0f / (float)n;
  float m = sums[f] * inv_n;
  float var = sqs[f] * inv_n - m * m;
  float y = gam[f] * (x[idx] - m) * rsqrtf(var + BN_EPS) + bet[f];
  x[idx] = tanhf(y);
}

// ---------------- score helpers ----------------
__global__ void dotw(const float* __restrict__ h, const float* __restrict__ w,
                     float* __restrict__ p, int n, int F) {
  int v = blockIdx.x * blockDim.x + threadIdx.x;
  if (v >= n) return;
  float acc = 0.0f;
  for (int f = 0; f < F; ++f) acc += h[(size_t)v * F + f] * w[f];
  p[v] = acc;
}
__global__ void score_edge_agg(const float* __restrict__ p, const int* __restrict__ srcv,
                               const int* __restrict__ dstv, float* __restrict__ a, int nE) {
  int e = blockIdx.x * blockDim.x + threadIdx.x;
  if (e >= nE) return;
  int s = srcv[e], d = dstv[e];
  if (s >= 0 && d >= 0) atomicAdd(&a[d], p[s]);
}
__global__ void score_final(const float* __restrict__ h, const float* __restrict__ wr,
                            const float* __restrict__ a, const float* __restrict__ bias,
                            float* __restrict__ s, int n, int F) {
  int v = blockIdx.x * blockDim.x + threadIdx.x;
  if (v >= n) return;
  float acc = bias[0] + a[v];
  for (int f = 0; f < F; ++f) acc += h[(size_t)v * F + f] * wr[f];
  s[v] = tanhf(acc);
}

// ---------------- per-graph top-K1: exact K-th largest via bitwise search ----------------
__global__ void topk_select(const float* __restrict__ s, int* __restrict__ perm) {
  int g = blockIdx.x;                       // one block per graph
  const float* sg = s + (size_t)g * NPG;
  __shared__ int red[256];
  unsigned T = 0;
  for (int bit = 31; bit >= 0; --bit) {
    unsigned cand = T | (1u << bit);
    int local = 0;
    for (int i = threadIdx.x; i < NPG; i += blockDim.x)
      if (f2u(sg[i]) >= cand) ++local;
    red[threadIdx.x] = local;
    __syncthreads();
    for (int st = 128; st > 0; st >>= 1) {
      if (threadIdx.x < st) red[threadIdx.x] += red[threadIdx.x + st];
      __syncthreads();
    }
    if (red[0] >= K1) T = cand;             // uniform decision
    __syncthreads();
  }
  if (threadIdx.x == 0) {                   // deterministic index-order compaction
    int cnt = 0;
    for (int i = 0; i < NPG && cnt < K1; ++i)
      if (f2u(sg[i]) > T) perm[g * K1 + cnt++] = g * NPG + i;
    for (int i = 0; i < NPG && cnt < K1; ++i)
      if (f2u(sg[i]) == T) perm[g * K1 + cnt++] = g * NPG + i;
  }
}

__global__ void pool_gather(const float* __restrict__ h, const float* __restrict__ s,
                            const int* __restrict__ perm, float* __restrict__ hp) {
  size_t idx = (size_t)blockIdx.x * blockDim.x + threadIdx.x;
  if (idx >= (size_t)N2 * 32) return;
  int v = (int)(idx / 32), f = (int)(idx % 32);
  int node = perm[v];
  hp[idx] = h[(size_t)node * 32 + f] * s[node];
}

__global__ void set_nidx(int* __restrict__ nidx, const int* __restrict__ perm) {
  int i = blockIdx.x * blockDim.x + threadIdx.x;
  if (i < N2) nidx[perm[i]] = i;
}
__global__ void remap_edges(const int* __restrict__ srcv, const int* __restrict__ dstv,
                            const int* __restrict__ nidx, int* __restrict__ s2,
                            int* __restrict__ d2) {
  int e = blockIdx.x * blockDim.x + threadIdx.x;
  if (e >= Eg) return;
  s2[e] = nidx[srcv[e]];
  d2[e] = nidx[dstv[e]];
}

// ---------------- final: per-graph argmax, out = h[argmax] * score ----------------
__global__ void final_argmax(const float* __restrict__ h, const float* __restrict__ sc,
                             float* __restrict__ out) {
  int g = blockIdx.x;
  __shared__ float bv[256];
  __shared__ int bi[256];
  float best = -INFINITY;
  int bidx = 0x7fffffff;
  for (int i = threadIdx.x; i < K1; i += blockDim.x) {
    float v = sc[g * K1 + i];
    if (v > best || (v == best && i < bidx)) { best = v; bidx = i; }
  }
  bv[threadIdx.x] = best;
  bi[threadIdx.x] = bidx;
  __syncthreads();
  for (int st = 128; st > 0; st >>= 1) {
    if (threadIdx.x < st) {
      if (bv[threadIdx.x + st] > bv[threadIdx.x] ||
          (bv[threadIdx.x + st] == bv[threadIdx.x] && bi[threadIdx.x + st] < bi[threadIdx.x])) {
        bv[threadIdx.x] = bv[threadIdx.x + st];
        bi[threadIdx.x] = bi[threadIdx.x + st];
      }
    }
    __syncthreads();
  }
  int node = g * K1 + bi[0];
  float val = bv[0];
  for (int f = threadIdx.x; f < 64; f += blockDim.x)
    out[g * 64 + f] = h[(size_t)node * 64 + f] * val;
}

inline unsigned cdiv(size_t a, unsigned b) { return (unsigned)((a + b - 1) / b); }
}  // namespace

extern "C" void kernel_launch(void* const* d_in, const int* in_sizes, int n_in,
                              void* d_out, int out_size, void* d_ws, size_t ws_size,
                              hipStream_t stream) {
  (void)in_sizes; (void)n_in; (void)out_size; (void)ws_size;
  const float* x    = (const float*)d_in[0];
  const int*   ei   = (const int*)d_in[1];
  const int*   srcv = ei;
  const int*   dstv = ei + Eg;
  const float* W1 = (const float*)d_in[3],  *b1 = (const float*)d_in[4];
  const float* W2 = (const float*)d_in[5],  *b2 = (const float*)d_in[6];
  const float* W3 = (const float*)d_in[7],  *b3 = (const float*)d_in[8];
  const float* W4 = (const float*)d_in[9],  *b4 = (const float*)d_in[10];
  const float* W5 = (const float*)d_in[11], *b5 = (const float*)d_in[12];
  const float* g1 = (const float*)d_in[13], *be1 = (const float*)d_in[14];
  const float* g2 = (const float*)d_in[15], *be2 = (const float*)d_in[16];
  const float* g3 = (const float*)d_in[17], *be3 = (const float*)d_in[18];
  const float* g4 = (const float*)d_in[19], *be4 = (const float*)d_in[20];
  const float* sWr1 = (const float*)d_in[21], *sWn1 = (const float*)d_in[22];
  const float* sbp1 = (const float*)d_in[23];
  const float* sWr2 = (const float*)d_in[24], *sWn2 = (const float*)d_in[25];
  const float* sbp2 = (const float*)d_in[26];
  float* out = (float*)d_out;

  // ---- carve workspace ----
  char* ws = (char*)d_ws;
  size_t off = 0;
  auto alloc = [&](size_t bytes) -> void* {
    void* p = ws + off;
    off = (off + bytes + 255) & ~(size_t)255;
    return p;
  };
  float* fA   = (float*)alloc((size_t)Nn * 32 * 4);  // temp / stage-2 features
  float* fB   = (float*)alloc((size_t)Nn * 32 * 4);  // stage-1 features / stage-2 temp
  float* deg  = (float*)alloc((size_t)Nn * 4);
  float* dinv = (float*)alloc((size_t)Nn * 4);
  float* pbuf = (float*)alloc((size_t)Nn * 4);
  float* sagg = (float*)alloc((size_t)Nn * 4);
  float* sbuf = (float*)alloc((size_t)Nn * 4);
  int*   perm = (int*)alloc((size_t)N2 * 4);
  int*   nidx = (int*)alloc((size_t)Nn * 4);
  int*   s2   = (int*)alloc((size_t)Eg * 4);
  int*   d2   = (int*)alloc((size_t)Eg * 4);
  float* stats = (float*)alloc(128 * 4);             // [sums(64) | sqs(64)]
  float* sums = stats, *sqs = stats + 64;

  const unsigned TB = 256;
  const unsigned eB = cdiv(Eg, TB);
  const unsigned nB = cdiv(Nn, TB);

  // ---- stage 1 degree norm ----
  fill_f<<<nB, TB, 0, stream>>>(deg, 1.0f, (size_t)Nn);
  deg_acc<<<eB, TB, 0, stream>>>(srcv, dstv, deg, Eg);
  rsq_k<<<nB, TB, 0, stream>>>(deg, dinv, Nn);

  auto gcn_bn_32 = [&](const float* Xin, int Kdim, const float* W, const float* b,
                       const float* gam, const float* bet) {
    if (Kdim == 7)
      gemm_wmma<7, 32><<<cdiv((size_t)(Nn / 16) * 2, 8), TB, 0, stream>>>(Xin, W, fA, Nn);
    else
      gemm_wmma<32, 32><<<cdiv((size_t)(Nn / 16) * 2, 8), TB, 0, stream>>>(Xin, W, fA, Nn);
    fill_f<<<cdiv((size_t)Nn * 32, TB), TB, 0, stream>>>(fB, 0.0f, (size_t)Nn * 32);
    edge_agg<32><<<cdiv((size_t)Eg * 32, TB), TB, 0, stream>>>(fA, srcv, dstv, dinv, fB, Eg);
    selfloop_bias<32><<<cdiv((size_t)Nn * 32, TB), TB, 0, stream>>>(fB, fA, dinv, b, Nn);
    fill_f<<<1, 128, 0, stream>>>(stats, 0.0f, 128);
    bn_stats<32><<<1024, TB, 0, stream>>>(fB, sums, sqs, Nn);
    bn_apply_tanh<32><<<cdiv((size_t)Nn * 32, TB), TB, 0, stream>>>(fB, sums, sqs, gam, bet, Nn);
  };

  gcn_bn_32(x,  7,  W1, b1, g1, be1);   // h1 in fB
  gcn_bn_32(fB, 32, W2, b2, g2, be2);   // h2 in fB
  gcn_bn_32(fB, 32, W3, b3, g3, be3);   // h3 in fB

  // ---- SAGPool 1 ----
  dotw<<<nB, TB, 0, stream>>>(fB, sWn1, pbuf, Nn, 32);
  fill_f<<<nB, TB, 0, stream>>>(sagg, 0.0f, (size_t)Nn);
  score_edge_agg<<<eB, TB, 0, stream>>>(pbuf, srcv, dstv, sagg, Eg);
  score_final<<<nB, TB, 0, stream>>>(fB, sWr1, sagg, sbp1, sbuf, Nn, 32);
  topk_select<<<Bg, TB, 0, stream>>>(sbuf, perm);
  pool_gather<<<cdiv((size_t)N2 * 32, TB), TB, 0, stream>>>(fB, sbuf, perm, fA);  // hp in fA
  fill_i<<<nB, TB, 0, stream>>>(nidx, -1, (size_t)Nn);
  set_nidx<<<cdiv(N2, TB), TB, 0, stream>>>(nidx, perm);
  remap_edges<<<eB, TB, 0, stream>>>(srcv, dstv, nidx, s2, d2);

  // ---- stage 2 degree norm (masked edges) ----
  fill_f<<<cdiv(N2, TB), TB, 0, stream>>>(deg, 1.0f, (size_t)N2);
  deg_acc<<<eB, TB, 0, stream>>>(s2, d2, deg, Eg);
  rsq_k<<<cdiv(N2, TB), TB, 0, stream>>>(deg, dinv, N2);

  // ---- GCN4 (32->64) + BN + tanh; features end in fA ----
  gemm_wmma<32, 64><<<cdiv((size_t)(N2 / 16) * 4, 8), TB, 0, stream>>>(fA, W4, fB, N2);
  fill_f<<<cdiv((size_t)N2 * 64, TB), TB, 0, stream>>>(fA, 0.0f, (size_t)N2 * 64);
  edge_agg<64><<<cdiv((size_t)Eg * 64, TB), TB, 0, stream>>>(fB, s2, d2, dinv, fA, Eg);
  selfloop_bias<64><<<cdiv((size_t)N2 * 64, TB), TB, 0, stream>>>(fA, fB, dinv, b4, N2);
  fill_f<<<1, 128, 0, stream>>>(stats, 0.0f, 128);
  bn_stats<64><<<512, TB, 0, stream>>>(fA, sums, sqs, N2);
  bn_apply_tanh<64><<<cdiv((size_t)N2 * 64, TB), TB, 0, stream>>>(fA, sums, sqs, g4, be4, N2);

  // ---- GCN5 (64->64), no BN; features end in fA ----
  gemm_wmma<64, 64><<<cdiv((size_t)(N2 / 16) * 4, 8), TB, 0, stream>>>(fA, W5, fB, N2);
  fill_f<<<cdiv((size_t)N2 * 64, TB), TB, 0, stream>>>(fA, 0.0f, (size_t)N2 * 64);
  edge_agg<64><<<cdiv((size_t)Eg * 64, TB), TB, 0, stream>>>(fB, s2, d2, dinv, fA, Eg);
  selfloop_bias<64><<<cdiv((size_t)N2 * 64, TB), TB, 0, stream>>>(fA, fB, dinv, b5, N2);

  // ---- SAGPool 2 (K2 = 1 -> per-graph argmax) ----
  dotw<<<cdiv(N2, TB), TB, 0, stream>>>(fA, sWn2, pbuf, N2, 64);
  fill_f<<<cdiv(N2, TB), TB, 0, stream>>>(sagg, 0.0f, (size_t)N2);
  score_edge_agg<<<eB, TB, 0, stream>>>(pbuf, s2, d2, sagg, Eg);
  score_final<<<cdiv(N2, TB), TB, 0, stream>>>(fA, sWr2, sagg, sbp2, sbuf, N2, 64);
  final_argmax<<<Bg, TB, 0, stream>>>(fA, sbuf, out);
}